// PRED_51857435132435
// MI455X (gfx1250) — compile-verified
//
#include <hip/hip_runtime.h>

typedef __attribute__((ext_vector_type(2))) float v2f;
typedef __attribute__((ext_vector_type(8))) float v8f;

#define B_SZ   256
#define T_SZ   1024
#define C_SZ   64
#define HID    32
#define CHUNK  256

// Padded LDS row widths (bank-conflict avoidance, 64 banks x 4B)
#define YPAD   260   // y_chunk row stride
#define GPAD   132   // gates row stride
#define HPAD   17    // h row stride (17 coprime 64 -> conflict-free scatter)

// ---------------------------------------------------------------------------
// Fast branch-free activations on CDNA5. V_TANH_F32 is a single TRANS op on
// gfx1250; sigmoid(x) = 0.5*tanh(0.5x)+0.5. Fallback: raw v_exp_f32/v_rcp_f32
// (full-range in hardware, no libm slow path, no EXEC divergence).
// ---------------------------------------------------------------------------
#if __has_builtin(__builtin_amdgcn_tanhf)
__device__ __forceinline__ float fast_tanh(float x) {
    return __builtin_amdgcn_tanhf(x);
}
__device__ __forceinline__ float fast_sigmoid(float x) {
    return __builtin_amdgcn_tanhf(0.5f * x) * 0.5f + 0.5f;
}
#else
__device__ __forceinline__ float fast_tanh(float x) {
    // tanh(x) = 1 - 2/(exp(2x)+1); exp(2x) = exp2(2*log2(e)*x)
    float e = __builtin_amdgcn_exp2f(2.8853900817779268f * x);
    return 1.f - 2.f * __builtin_amdgcn_rcpf(e + 1.f);
}
__device__ __forceinline__ float fast_sigmoid(float x) {
    float e = __builtin_amdgcn_exp2f(-1.4426950408889634f * x);
    return __builtin_amdgcn_rcpf(1.f + e);
}
#endif

// ---------------------------------------------------------------------------
// Phase 1: 1x1 conv = per-pixel channel dot product. Bandwidth-bound (64 MB
// of x @ 23.3 TB/s ~ 2.9 us). One block per batch row; float4 per thread.
// ---------------------------------------------------------------------------
__global__ __launch_bounds__(256) void conv1x1_kernel(
    const float* __restrict__ x,       // [B][C][T]
    const float* __restrict__ conv_w,  // [C]
    const float* __restrict__ conv_b,  // [1]
    float* __restrict__ y)             // [B][T]
{
    const int b = blockIdx.x;
    const int t = threadIdx.x * 4;
    const float* xb = x + (size_t)b * (C_SZ * T_SZ) + t;

    float4 acc = make_float4(0.f, 0.f, 0.f, 0.f);
#pragma unroll 8
    for (int c = 0; c < C_SZ; ++c) {
        const float4 v = *(const float4*)(xb + (size_t)c * T_SZ);
        const float w = conv_w[c];
        acc.x += w * v.x; acc.y += w * v.y;
        acc.z += w * v.z; acc.w += w * v.w;
    }
    const float bb = conv_b[0];
    acc.x += bb; acc.y += bb; acc.z += bb; acc.w += bb;
    *(float4*)(y + (size_t)b * T_SZ + t) = acc;
}

// ---------------------------------------------------------------------------
// Phase 2: LSTM scan. 16 workgroups, each owns 16 batch rows.
// Per step: gates(16x128) = h(16x32) @ w_hh^T(32x128) + y[b,t]*w_ih + bias
// via 8 waves x (8 chained V_WMMA_F32_16X16X4_F32), fp32-exact accumulation.
// ---------------------------------------------------------------------------
__global__ __launch_bounds__(256) void lstm_kernel(
    const float* __restrict__ y,        // [B][T]
    const float* __restrict__ w_ih,     // [4H][1] -> flat [128]
    const float* __restrict__ w_hh,     // [4H][H] = [128][32]
    const float* __restrict__ b_ih,     // [128]
    const float* __restrict__ b_hh,     // [128]
    const float* __restrict__ dense_w,  // [32]
    const float* __restrict__ dense_b,  // [1]
    float* __restrict__ out)            // [B]
{
    __shared__ float y_chunk[16][YPAD];     // staged conv outputs (1 chunk)
    __shared__ float gates[16][GPAD];       // pre-activation gates
    __shared__ float h_lds[HID][HPAD];      // h, K-major for A-fragment

    const int tid  = threadIdx.x;
    const int wave = tid >> 5;          // 0..7 : N-tile (gate column block)
    const int lane = tid & 31;
    const int hl   = lane >> 4;         // lane half (0/1)
    const int ml   = lane & 15;
    const int col  = wave * 16 + ml;    // gate column 0..127
    const int b0   = blockIdx.x * 16;   // batch tile base

    // Per-lane constants for accumulator init (input projection + bias)
    const float wih  = w_ih[col];
    const float bias = b_ih[col] + b_hh[col];

    // B-fragments of w_hh^T, constant across steps, kept in registers.
    // B 4x16 layout: VGPR0 = K0 | K2 , VGPR1 = K1 | K3 (lane halves), N = ml.
    float breg[8][2];
#pragma unroll
    for (int p = 0; p < 8; ++p) {
#pragma unroll
        for (int v = 0; v < 2; ++v) {
            const int k = 4 * p + v + 2 * hl;
            breg[p][v] = w_hh[col * HID + k];   // B[k][n] = w_hh[col][k]
        }
    }

    // Cell-update element ownership: 512 elems (16 batch x 32 hid), 2/thread.
    const int e0 = tid, e1 = tid + 256;
    const int m0 = e0 >> 5, j0 = e0 & 31;
    const int m1 = e1 >> 5, j1 = e1 & 31;
    float c0 = 0.f, c1 = 0.f;

    // h = 0 (padded array: zero all HID*HPAD entries)
    for (int i = tid; i < HID * HPAD; i += 256) ((float*)h_lds)[i] = 0.f;
    __syncthreads();

    for (int t = 0; t < T_SZ; ++t) {
        const int tt = t & (CHUNK - 1);
        if (tt == 0) {
            __syncthreads();
            // Refill y_chunk for steps [t, t+CHUNK). Coalesced: 16 threads
            // per batch row, 4x float4 each.
            const int mrow = tid >> 4;
            const int l16  = tid & 15;
            const float* src = y + (size_t)(b0 + mrow) * T_SZ + t + l16 * 16;
#pragma unroll
            for (int q = 0; q < 4; ++q) {
                const float4 v = *(const float4*)(src + q * 4);
                *(float4*)&y_chunk[mrow][l16 * 16 + q * 4] = v;
            }
            __syncthreads();
        }

        // A-fragment (h tile 16x32), f32 A 16x4 layout per K-chunk:
        // VGPR 2p = K(4p)|K(4p+2), VGPR 2p+1 = K(4p+1)|K(4p+3), M = ml.
        float a[16];
#pragma unroll
        for (int p = 0; p < 8; ++p) {
            const int kb = 4 * p + 2 * hl;
            a[2 * p]     = h_lds[kb][ml];
            a[2 * p + 1] = h_lds[kb + 1][ml];
        }

        // Accumulator init = y[b,t] * w_ih[col] + bias, in C/D layout:
        // VGPR r holds M = r + 8*hl, N = ml.
        v8f acc;
#pragma unroll
        for (int r = 0; r < 8; ++r) {
            acc[r] = y_chunk[r + 8 * hl][tt] * wih + bias;
        }

        // K=32 contraction: 8 chained fp32 WMMAs (exact accumulation).
#pragma unroll
        for (int p = 0; p < 8; ++p) {
            v2f av; av[0] = a[2 * p];     av[1] = a[2 * p + 1];
            v2f bv; bv[0] = breg[p][0];   bv[1] = breg[p][1];
            acc = __builtin_amdgcn_wmma_f32_16x16x4_f32(
                false, av, false, bv, (short)0, acc, false, false);
        }

        // Scatter gate pre-activations to LDS (bank-disjoint lane halves).
#pragma unroll
        for (int r = 0; r < 8; ++r) {
            gates[r + 8 * hl][col] = acc[r];
        }
        __syncthreads();

        // Elementwise cell update, 2 (m, j) elements per thread.
        // Activations are single hardware TRANS ops -> short serial chain.
        {
            const float i = fast_sigmoid(gates[m0][j0]);
            const float f = fast_sigmoid(gates[m0][HID + j0]);
            const float g = fast_tanh   (gates[m0][2 * HID + j0]);
            const float o = fast_sigmoid(gates[m0][3 * HID + j0]);
            c0 = f * c0 + i * g;
            h_lds[j0][m0] = o * fast_tanh(c0);
        }
        {
            const float i = fast_sigmoid(gates[m1][j1]);
            const float f = fast_sigmoid(gates[m1][HID + j1]);
            const float g = fast_tanh   (gates[m1][2 * HID + j1]);
            const float o = fast_sigmoid(gates[m1][3 * HID + j1]);
            c1 = f * c1 + i * g;
            h_lds[j1][m1] = o * fast_tanh(c1);
        }
        __syncthreads();
    }

    // Final dense: out[b] = h_last . dense_w + dense_b
    if (tid < 16) {
        float s = 0.f;
#pragma unroll
        for (int j = 0; j < HID; ++j) s += h_lds[j][tid] * dense_w[j];
        out[b0 + tid] = s + dense_b[0];
    }
}

// ---------------------------------------------------------------------------
extern "C" void kernel_launch(void* const* d_in, const int* in_sizes, int n_in,
                              void* d_out, int out_size, void* d_ws, size_t ws_size,
                              hipStream_t stream) {
    const float* x       = (const float*)d_in[0];
    const float* conv_w  = (const float*)d_in[1];
    const float* conv_b  = (const float*)d_in[2];
    const float* w_ih    = (const float*)d_in[3];
    const float* w_hh    = (const float*)d_in[4];
    const float* b_ih    = (const float*)d_in[5];
    const float* b_hh    = (const float*)d_in[6];
    const float* dense_w = (const float*)d_in[7];
    const float* dense_b = (const float*)d_in[8];

    float* y = (float*)d_ws;  // [B][T] = 1 MB scratch

    conv1x1_kernel<<<B_SZ, 256, 0, stream>>>(x, conv_w, conv_b, y);
    lstm_kernel<<<16, 256, 0, stream>>>(y, w_ih, w_hh, b_ih, b_hh,
                                        dense_w, dense_b, (float*)d_out);
}